// LTCCell_45294725104193
// MI455X (gfx1250) — compile-verified
//
#include <hip/hip_runtime.h>

// ---------------------------------------------------------------------------
// LTC cell (liquid time-constant RNN step), MI455X / gfx1250.
//
// Core op: w_num[b,n] = sum_p w[p,n]*erev[p,n]*sigmoid(sigma[p,n]*(v[b,p]-mu[p,n]))
// The sigmoid couples (b,p) and (p,n) inside the nonlinearity -> no matmul
// factorization exists, so WMMA cannot express this. The kernel is
// transcendental/VALU bound (~400M gate evals, 2 transcendentals each).
// CDNA5-specific paths used: async global->LDS staging (ASYNCcnt),
// LDS broadcast loads, wave32 lane-per-column mapping, b128 packed params.
// ---------------------------------------------------------------------------

#define UNITS   512
#define SENSORY 256
#define BATCH   256
#define UNFOLDS 6
#define B_TILE  2
#define THREADS 512
#define L2E     1.44269504088896340736f

#if defined(__has_builtin)
#if __has_builtin(__builtin_amdgcn_global_load_async_to_lds_b32)
#define LTC_ASYNC 1
#endif
#endif

typedef __attribute__((address_space(1))) int ltc_gint;  // global addrspace
typedef __attribute__((address_space(3))) int ltc_lint;  // LDS addrspace

__device__ __forceinline__ void g2lds(const float* __restrict__ g, float* l) {
#if defined(LTC_ASYNC)
  __builtin_amdgcn_global_load_async_to_lds_b32(
      (ltc_gint*)g, (ltc_lint*)l, /*offset=*/0, /*cpol=*/0);
#else
  *l = *g;
#endif
}

__device__ __forceinline__ void wait_lds_copies() {
#if defined(LTC_ASYNC)
#if __has_builtin(__builtin_amdgcn_s_wait_asynccnt)
  __builtin_amdgcn_s_wait_asynccnt(0);
#else
  asm volatile("s_wait_asynccnt 0" ::: "memory");
#endif
#endif
}

// Pack parameters for the inner loop:
//   .x = sigma*log2e          (gate slope, pre-scaled for v_exp_f32/exp2)
//   .y = sigma*log2e*mu       (gate offset)
//   .z = w                    (denominator weight)
//   .w = w*erev               (numerator weight)
// Gate: m = .y - .x*v ; u = exp2(m) = exp(-sigma*(v-mu)) ; s = 1/(1+u)
__global__ __launch_bounds__(256) void ltc_pack(
    const float* __restrict__ sigma, const float* __restrict__ mu,
    const float* __restrict__ w,     const float* __restrict__ erev,
    const float* __restrict__ ssig,  const float* __restrict__ smu,
    const float* __restrict__ sw,    const float* __restrict__ serev,
    float4* __restrict__ packR, float4* __restrict__ packS)
{
  int i = blockIdx.x * 256 + threadIdx.x;
  if (i < UNITS * UNITS) {
    float a = sigma[i] * L2E;
    float wi = w[i];
    packR[i] = make_float4(a, a * mu[i], wi, wi * erev[i]);
  }
  if (i < SENSORY * UNITS) {
    float a = ssig[i] * L2E;
    float wi = sw[i];
    packS[i] = make_float4(a, a * smu[i], wi, wi * serev[i]);
  }
}

__global__ __launch_bounds__(THREADS) void ltc_main(
    const float* __restrict__ x,        const float* __restrict__ state,
    const float* __restrict__ gleak,    const float* __restrict__ vleak,
    const float* __restrict__ cm,
    const float* __restrict__ input_w,  const float* __restrict__ input_b,
    const float* __restrict__ output_w, const float* __restrict__ output_b,
    const float4* __restrict__ packR,   const float4* __restrict__ packS,
    float* __restrict__ out)  // [0, B*N): outputs ; [B*N, 2*B*N): v_pre
{
  __shared__ float v_sh[B_TILE][UNITS];     // v_pre tile, lives here all 6 unfolds
  __shared__ float xw_sh[B_TILE][SENSORY];  // input_w*x + input_b tile

  const int t  = threadIdx.x;               // this thread owns column n = t
  const int b0 = blockIdx.x * B_TILE;

  // ---- Stage state and x rows into LDS (async global->LDS on CDNA5) ----
  for (int i = t; i < B_TILE * UNITS; i += THREADS)
    g2lds(&state[b0 * UNITS + i], &v_sh[0][0] + i);
  for (int i = t; i < B_TILE * SENSORY; i += THREADS)
    g2lds(&x[b0 * SENSORY + i], &xw_sh[0][0] + i);
  wait_lds_copies();
  __syncthreads();

  // inputs = x*input_w + input_b, in place
  for (int i = t; i < B_TILE * SENSORY; i += THREADS) {
    int d = i & (SENSORY - 1);
    (&xw_sh[0][0])[i] = fmaf((&xw_sh[0][0])[i], input_w[d], input_b[d]);
  }
  __syncthreads();

  // ---- Sensory accumulation (computed once, reused every unfold) ----
  float sens_n[B_TILE], sens_d[B_TILE];
#pragma unroll
  for (int b = 0; b < B_TILE; ++b) { sens_n[b] = 0.f; sens_d[b] = 0.f; }

#pragma unroll 2
  for (int d = 0; d < SENSORY; ++d) {
    float4 pk = packS[d * UNITS + t];       // global_load_b128, L2-resident
#pragma unroll
    for (int b = 0; b < B_TILE; ++b) {
      float m = fmaf(-pk.x, xw_sh[b][d], pk.y);
      float u = __builtin_amdgcn_exp2f(m);          // v_exp_f32
      float s = __builtin_amdgcn_rcpf(1.0f + u);    // v_rcp_f32
      sens_d[b] = fmaf(pk.z, s, sens_d[b]);
      sens_n[b] = fmaf(pk.w, s, sens_n[b]);
    }
  }

  const float gl   = gleak[t];
  const float glvl = gl * vleak[t];
  const float cmt  = cm[t] * (6.0f / 1.001f);   // cm / ((t+eps)/unfolds)

  // ---- 6 ODE unfolds, v_pre held in LDS, only workgroup barriers ----
  for (int unf = 0; unf < UNFOLDS; ++unf) {
    float an[B_TILE], ad[B_TILE];
#pragma unroll
    for (int b = 0; b < B_TILE; ++b) { an[b] = sens_n[b]; ad[b] = sens_d[b]; }

#pragma unroll 2
    for (int p = 0; p < UNITS; ++p) {
      float4 pk = packR[p * UNITS + t];     // b128, streams from L2
#pragma unroll
      for (int b = 0; b < B_TILE; ++b) {
        float vb = v_sh[b][p];              // wave-uniform addr -> LDS broadcast
        float m = fmaf(-pk.x, vb, pk.y);
        float u = __builtin_amdgcn_exp2f(m);
        float s = __builtin_amdgcn_rcpf(1.0f + u);
        ad[b] = fmaf(pk.z, s, ad[b]);
        an[b] = fmaf(pk.w, s, an[b]);
      }
    }

    float vn[B_TILE];
#pragma unroll
    for (int b = 0; b < B_TILE; ++b) {
      float vold = v_sh[b][t];
      float num = fmaf(cmt, vold, glvl) + an[b];
      float den = cmt + gl + ad[b];
      vn[b] = num / (den + 1e-8f);          // exact div: only 12 per thread
    }
    __syncthreads();                         // everyone done reading old v
#pragma unroll
    for (int b = 0; b < B_TILE; ++b) v_sh[b][t] = vn[b];
    __syncthreads();                         // new v visible
  }

  // ---- Emit (outputs, v_pre) ----
  const float ow = output_w[t], ob = output_b[t];
#pragma unroll
  for (int b = 0; b < B_TILE; ++b) {
    float v = v_sh[b][t];
    out[(b0 + b) * UNITS + t]                 = fmaf(v, ow, ob);
    out[BATCH * UNITS + (b0 + b) * UNITS + t] = v;
  }
}

extern "C" void kernel_launch(void* const* d_in, const int* in_sizes, int n_in,
                              void* d_out, int out_size, void* d_ws, size_t ws_size,
                              hipStream_t stream) {
  const float* x        = (const float*)d_in[0];
  const float* state    = (const float*)d_in[1];
  const float* gleak    = (const float*)d_in[2];
  const float* vleak    = (const float*)d_in[3];
  const float* cm       = (const float*)d_in[4];
  const float* sigma    = (const float*)d_in[5];
  const float* mu       = (const float*)d_in[6];
  const float* w        = (const float*)d_in[7];
  const float* erev     = (const float*)d_in[8];
  const float* ssig     = (const float*)d_in[9];
  const float* smu      = (const float*)d_in[10];
  const float* sw       = (const float*)d_in[11];
  const float* serev    = (const float*)d_in[12];
  const float* input_w  = (const float*)d_in[13];
  const float* input_b  = (const float*)d_in[14];
  const float* output_w = (const float*)d_in[15];
  const float* output_b = (const float*)d_in[16];

  float* out = (float*)d_out;

  float4* packR = (float4*)d_ws;                                       // 4 MB
  float4* packS = (float4*)((char*)d_ws + (size_t)UNITS * UNITS * 16); // 2 MB

  ltc_pack<<<(UNITS * UNITS + 255) / 256, 256, 0, stream>>>(
      sigma, mu, w, erev, ssig, smu, sw, serev, packR, packS);

  ltc_main<<<BATCH / B_TILE, THREADS, 0, stream>>>(
      x, state, gleak, vleak, cm, input_w, input_b, output_w, output_b,
      packR, packS, out);
}